// ginn_autoencoder_skip_28681791603394
// MI455X (gfx1250) — compile-verified
//
#include <hip/hip_runtime.h>
#include <hip/hip_bf16.h>
#include <math.h>

// ---------------------------------------------------------------------------
// GINN autoencoder w/ skip for MI455X (gfx1250, wave32).
// GEMMs via V_WMMA_F32_16X16X4_F32 (full fp32 precision; GEMM flops are
// negligible vs. the ~2 GB edge-aggregation traffic, which is the real
// roofline limit at 23.3 TB/s). Aggregation via global_atomic_add_f32.
// ---------------------------------------------------------------------------

#define N_NODES 50000
#define N_EDGES 800000
#define IN_F    64
#define H_F     128

typedef float v2f __attribute__((ext_vector_type(2)));
typedef float v8f __attribute__((ext_vector_type(8)));

// ----------------------------- WMMA GEMM -----------------------------------
// C[M,N] = (A[M,K] @ B[K,N]) * rowScale[M]   (row-major A, B, C)
// One wave computes one 16x16 tile. 8 waves (256 threads) per block.
// fp32 WMMA layouts (CDNA5 ISA 7.12.2):
//   A 16x4 : lane r (r<16) row=r ; v0 = K0 (lanes 0-15) / K2 (lanes 16-31),
//            v1 = K1 / K3
//   B 4x16 : mirrored, lanes index N
//   C 16x16: vgpr i -> row = i + 8*(lane>=16), col = lane&15
template <int K, int N>
__global__ void __launch_bounds__(256)
gemm_rowscale_wmma(const float* __restrict__ A, const float* __restrict__ B,
                   const float* __restrict__ rowScale, float* __restrict__ C,
                   int totalTiles)
{
    const int lane = threadIdx.x & 31;
    const int wave = threadIdx.x >> 5;
    const int tile = blockIdx.x * 8 + wave;
    if (tile >= totalTiles) return;          // wave-uniform: EXEC all-1s inside

    constexpr int NT = N >> 4;               // tiles along N
    const int tm = tile / NT;
    const int tn = tile - tm * NT;
    const int r    = lane & 15;
    const int half = lane >> 4;              // 0: lanes 0-15, 1: lanes 16-31

    const float* __restrict__ Arow = A + (size_t)(tm * 16 + r) * K;
    const float* __restrict__ Bcol = B + (tn * 16 + r);

    v8f acc = {};
#pragma unroll
    for (int kc = 0; kc < K; kc += 4) {
        v2f a, b;
        a.x = Arow[kc + 2 * half + 0];
        a.y = Arow[kc + 2 * half + 1];
        b.x = Bcol[(size_t)(kc + 2 * half + 0) * N];
        b.y = Bcol[(size_t)(kc + 2 * half + 1) * N];
        acc = __builtin_amdgcn_wmma_f32_16x16x4_f32(
            /*neg_a=*/false, a, /*neg_b=*/false, b,
            /*c_mod=*/(short)0, acc, /*reuse_a=*/false, /*reuse_b=*/false);
    }

#pragma unroll
    for (int i = 0; i < 8; ++i) {
        const int row = tm * 16 + i + half * 8;
        const int col = tn * 16 + r;
        C[(size_t)row * N + col] = acc[i] * rowScale[row];
    }
}

// ------------------------- edge aggregation --------------------------------
// agg[dst[e]*F + f] += val[src[e]*F + f]; one thread handles a float4 quad.
// LOGQ = log2(F/4).
template <int LOGQ>
__global__ void __launch_bounds__(256)
seg_sum_atomic(const float* __restrict__ val, const int* __restrict__ src,
               const int* __restrict__ dst, float* __restrict__ agg, int total)
{
    const int tid = blockIdx.x * blockDim.x + threadIdx.x;
    if (tid >= total) return;
    const int e = tid >> LOGQ;
    const int q = tid & ((1 << LOGQ) - 1);
    const int s = src[e];
    const int d = dst[e];
    const float4 v = *(const float4*)(val + ((size_t)s << (LOGQ + 2)) + q * 4);
    float* out = agg + ((size_t)d << (LOGQ + 2)) + q * 4;
    atomicAdd(out + 0, v.x);
    atomicAdd(out + 1, v.y);
    atomicAdd(out + 2, v.z);
    atomicAdd(out + 3, v.w);
}

// --------------------------- pointwise ops ---------------------------------
__global__ void __launch_bounds__(256)
ew_mul(const float* __restrict__ a, const float* __restrict__ b,
       float* __restrict__ o, int n)
{
    const int i = blockIdx.x * blockDim.x + threadIdx.x;
    if (i < n) o[i] = a[i] * b[i];
}

__global__ void __launch_bounds__(256)
ew_zero(float* __restrict__ o, int n)
{
    const int i = blockIdx.x * blockDim.x + threadIdx.x;
    if (i < n) o[i] = 0.0f;
}

// o[i] = relu(agg[i] * norm[i>>LOGF] + bias[i & (F-1)])
template <int LOGF>
__global__ void __launch_bounds__(256)
bias_relu(const float* __restrict__ agg, const float* __restrict__ norm,
          const float* __restrict__ bias, float* __restrict__ o, int n)
{
    const int i = blockIdx.x * blockDim.x + threadIdx.x;
    if (i >= n) return;
    const int row = i >> LOGF;
    const int col = i & ((1 << LOGF) - 1);
    const float h = agg[i] * norm[row] + bias[col];
    o[i] = fmaxf(h, 0.0f);
}

// out = sigmoid(ha*ng + bh + sa*nf + bs), F = 64
__global__ void __launch_bounds__(256)
final_sigmoid(const float* __restrict__ ha, const float* __restrict__ sa,
              const float* __restrict__ ng, const float* __restrict__ nf,
              const float* __restrict__ bh, const float* __restrict__ bs,
              float* __restrict__ out, int n)
{
    const int i = blockIdx.x * blockDim.x + threadIdx.x;
    if (i >= n) return;
    const int row = i >> 6;
    const int col = i & 63;
    const float z = ha[i] * ng[row] + bh[col] + sa[i] * nf[row] + bs[col];
    out[i] = 1.0f / (1.0f + expf(-z));
}

// ---------------------------------------------------------------------------
extern "C" void kernel_launch(void* const* d_in, const int* in_sizes, int n_in,
                              void* d_out, int out_size, void* d_ws, size_t ws_size,
                              hipStream_t stream)
{
    (void)in_sizes; (void)n_in; (void)out_size; (void)ws_size;

    const float* features = (const float*)d_in[0];   // [50000,64]
    const float* mask     = (const float*)d_in[1];   // [50000,64]
    const float* norm_g   = (const float*)d_in[2];   // [50000]
    const float* norm_f   = (const float*)d_in[3];   // [50000]
    const int*   src_g    = (const int*)  d_in[4];   // [800000]
    const int*   dst_g    = (const int*)  d_in[5];
    const int*   src_f    = (const int*)  d_in[6];
    const int*   dst_f    = (const int*)  d_in[7];
    const float* W1       = (const float*)d_in[8];   // [64,128]
    const float* b1       = (const float*)d_in[9];   // [128]
    const float* wh       = (const float*)d_in[10];  // [128,64]
    const float* bh       = (const float*)d_in[11];  // [64]
    const float* ws       = (const float*)d_in[12];  // [64,64]
    const float* bs       = (const float*)d_in[13];  // [64]
    float* out = (float*)d_out;                      // [50000,64]

    // workspace layout (floats); regions reused once their producer/consumer
    // pair has finished (launches are sequential on `stream`).
    float* W = (float*)d_ws;
    float* X     = W;                           // 3.2M  : features*mask
    float* HPRE  = W + 3200000;                 // 6.4M  : (x@W1)*ng, then relu'd h
    float* HAGG  = W + 9600000;                 // 6.4M  : segment sum of HPRE
    float* HBPRE = W + 9600000;                 // 3.2M  : reuses HAGG region
    float* SBPRE = W + 12800000;                // 3.2M  : reuses HAGG region
    float* HBAGG = W + 16000000;                // 3.2M
    float* SBAGG = W + 19200000;                // 3.2M

    const int nX  = N_NODES * IN_F;             // 3,200,000
    const int nH  = N_NODES * H_F;              // 6,400,000
    auto cdiv = [](int a, int b) { return (a + b - 1) / b; };

    // 1) x = features * mask
    ew_mul<<<cdiv(nX, 256), 256, 0, stream>>>(features, mask, X, nX);

    // 2) zero h aggregation buffer
    ew_zero<<<cdiv(nH, 256), 256, 0, stream>>>(HAGG, nH);

    // 3) h_pre = (x @ W1) * norm_g          [50000,128]
    {
        const int tiles = (N_NODES / 16) * (H_F / 16);      // 25000
        gemm_rowscale_wmma<IN_F, H_F><<<cdiv(tiles, 8), 256, 0, stream>>>(
            X, W1, norm_g, HPRE, tiles);
    }

    // 4) HAGG = segment_sum(HPRE[src_g], dst_g)            F=128 -> LOGQ=5
    {
        const int total = N_EDGES * (H_F / 4);              // 25.6M
        seg_sum_atomic<5><<<cdiv(total, 256), 256, 0, stream>>>(
            HPRE, src_g, dst_g, HAGG, total);
    }

    // 5) h = relu(HAGG*norm_g + b1)  (write into HPRE region; HPRE consumed)
    bias_relu<7><<<cdiv(nH, 256), 256, 0, stream>>>(HAGG, norm_g, b1, HPRE, nH);

    // 6) zero hb/sb aggregation buffers (contiguous 6.4M floats)
    ew_zero<<<cdiv(nH, 256), 256, 0, stream>>>(HBAGG, nH);

    // 7) hb_pre = (h @ wh) * norm_g         [50000,64]  (HAGG region now free)
    {
        const int tiles = (N_NODES / 16) * (IN_F / 16);     // 12500
        gemm_rowscale_wmma<H_F, IN_F><<<cdiv(tiles, 8), 256, 0, stream>>>(
            HPRE, wh, norm_g, HBPRE, tiles);
    }

    // 8) sb_pre = (x @ ws) * norm_f         [50000,64]
    {
        const int tiles = (N_NODES / 16) * (IN_F / 16);     // 12500
        gemm_rowscale_wmma<IN_F, IN_F><<<cdiv(tiles, 8), 256, 0, stream>>>(
            X, ws, norm_f, SBPRE, tiles);
    }

    // 9) HBAGG = segment_sum(HBPRE[src_g], dst_g)          F=64 -> LOGQ=4
    {
        const int total = N_EDGES * (IN_F / 4);             // 12.8M
        seg_sum_atomic<4><<<cdiv(total, 256), 256, 0, stream>>>(
            HBPRE, src_g, dst_g, HBAGG, total);
        // 10) SBAGG = segment_sum(SBPRE[src_f], dst_f)
        seg_sum_atomic<4><<<cdiv(total, 256), 256, 0, stream>>>(
            SBPRE, src_f, dst_f, SBAGG, total);
    }

    // 11) out = sigmoid(HBAGG*norm_g + bh + SBAGG*norm_f + bs)
    final_sigmoid<<<cdiv(nX, 256), 256, 0, stream>>>(
        HBAGG, SBAGG, norm_g, norm_f, bh, bs, out, nX);
}